// GATLayer_670014898392
// MI455X (gfx1250) — compile-verified
//
#include <hip/hip_runtime.h>

// ---------------------------------------------------------------------------
// GAT layer for MI455X (gfx1250).
//  - GEMM xh = x @ W uses V_WMMA_F32_16X16X4_F32 (exact fp32 on matrix pipe).
//  - x-tile staged into LDS with GLOBAL_LOAD_ASYNC_TO_LDS_B128 (ASYNCcnt),
//    bypassing VGPRs; s_wait_asynccnt 0 + barrier before consumption.
//  - Edge phase is HBM/L2-bandwidth bound (~1.7 GB scatter) -> wave-per-edge
//    streaming with fp32 global atomics (global_atomic_add_f32 /
//    global_atomic_max_num_f32).
// ---------------------------------------------------------------------------

typedef __attribute__((ext_vector_type(2))) float v2f;
typedef __attribute__((ext_vector_type(8))) float v8f;

#define IN_CH  256
#define HEADS  8
#define CPH    32
#define SLOPE  0.2f
#define AS_STRIDE 258   // 256 + 2: row->row bank step = 2 banks, keeps 8B align

// ------------------------- GEMM: xh = x @ W --------------------------------
// Block: 512 threads = 16 waves. Block handles 16 node-rows x all 256 cols.
// Wave w computes the 16x16 output tile for columns [16w, 16w+16).
__global__ __launch_bounds__(512)
void gat_gemm_wmma(const float* __restrict__ x, const float* __restrict__ W,
                   float* __restrict__ xh, int n)
{
    __shared__ float As[16 * AS_STRIDE];   // single LDS object -> offset 0

    const int tid  = threadIdx.x;
    const int row0 = blockIdx.x * 16;

    if (row0 + 16 <= n) {
        // Async direct-to-LDS staging: 16x256 floats = 16KB.
        // 512 threads x 2 passes x 16B (b128) each.
#pragma unroll
        for (int pass = 0; pass < 2; ++pass) {
            const int c  = tid + pass * 512;     // 16B-chunk index, 0..1023
            const int r  = c >> 6;               // 64 chunks per 256-float row
            const int k  = (c & 63) << 2;        // float offset within row
            const unsigned lds_off =
                (unsigned)((r * AS_STRIDE + k) * sizeof(float));
            const float* gp = x + (size_t)(row0 + r) * IN_CH + k;
            asm volatile("global_load_async_to_lds_b128 %0, %1, off"
                         :: "v"(lds_off), "v"(gp) : "memory");
        }
        asm volatile("s_wait_asynccnt 0x0" ::: "memory");
    } else {
        // Tail fallback (not taken for n % 16 == 0): zero-padded sync copy.
        for (int i = tid; i < 16 * IN_CH; i += 512) {
            const int r = i >> 8;
            const int k = i & 255;
            const int gr = row0 + r;
            As[r * AS_STRIDE + k] = (gr < n) ? x[(size_t)gr * IN_CH + k] : 0.0f;
        }
    }
    __syncthreads();

    const int wave = tid >> 5;
    const int lane = tid & 31;
    const int col0 = wave * 16;
    const int mn   = lane & 15;          // M for A-frag, N for B-frag
    const int khi  = (lane >> 4) << 1;   // lanes 16-31 hold K+2,K+3

    v8f acc = {};
    const float* Wb = W + col0 + mn;                 // column slab of W
    const float* Ab = &As[mn * AS_STRIDE + khi];

#pragma unroll 8
    for (int k = 0; k < IN_CH; k += 4) {
        v2f a, b;
        a.x = Ab[k];                       // A(mn, k+khi)
        a.y = Ab[k + 1];                   // A(mn, k+khi+1)
        b.x = Wb[(size_t)(k + khi) * IN_CH];       // B(k+khi,   col0+mn)
        b.y = Wb[(size_t)(k + khi + 1) * IN_CH];   // B(k+khi+1, col0+mn)
#if __has_builtin(__builtin_amdgcn_wmma_f32_16x16x4_f32)
        acc = __builtin_amdgcn_wmma_f32_16x16x4_f32(
                  false, a, false, b, (short)0, acc, false, false);
#else
        // (compile-survival fallback only; WMMA path is the intended one)
        acc[0] += a.x * b.x + a.y * b.y;
#endif
    }

    // C/D layout: lanes 0-15 -> N=lane, M=vgpr i; lanes 16-31 -> M=8+i.
    const int nc = lane & 15;
    const int mb = (lane >> 4) * 8;
#pragma unroll
    for (int i = 0; i < 8; ++i) {
        const int r = row0 + mb + i;
        if (r < n) xh[(size_t)r * IN_CH + col0 + nc] = acc[i];
    }
}

// ----------------- per-node attention logits (a_src, a_dst) ----------------
__global__ __launch_bounds__(256)
void gat_logits(const float* __restrict__ xh,
                const float* __restrict__ att_src,
                const float* __restrict__ att_dst,
                float* __restrict__ a_src, float* __restrict__ a_dst, int n)
{
    const int idx = blockIdx.x * blockDim.x + threadIdx.x;   // (node, head)
    if (idx >= n * HEADS) return;
    const int node = idx >> 3;
    const int h    = idx & 7;
    const float4* xv = (const float4*)(xh + (size_t)node * IN_CH + h * CPH);
    const float4* sv = (const float4*)(att_src + h * CPH);
    const float4* dv = (const float4*)(att_dst + h * CPH);
    float ss = 0.0f, dd = 0.0f;
#pragma unroll
    for (int i = 0; i < 8; ++i) {
        const float4 v = xv[i];
        const float4 s = sv[i];
        const float4 d = dv[i];
        ss += v.x * s.x + v.y * s.y + v.z * s.z + v.w * s.w;
        dd += v.x * d.x + v.y * d.y + v.z * d.z + v.w * d.w;
    }
    a_src[idx] = ss;
    a_dst[idx] = dd;
}

// ------------------- init: out=0, m=-inf, denom=0 --------------------------
__global__ __launch_bounds__(256)
void gat_init(float* __restrict__ out, float* __restrict__ m,
              float* __restrict__ denom, int n)
{
    const int i = blockIdx.x * blockDim.x + threadIdx.x;
    if (i < n * IN_CH) out[i] = 0.0f;
    if (i < n * HEADS) {
        m[i] = -__builtin_inff();
        denom[i] = 0.0f;
    }
}

// ------ per-edge leaky-relu logits + segment max (float atomic max) --------
__global__ __launch_bounds__(256)
void gat_edge_max(const int* __restrict__ ei,
                  const float* __restrict__ a_src,
                  const float* __restrict__ a_dst,
                  float* __restrict__ alpha, float* __restrict__ m,
                  int E, int total)
{
    const int e = blockIdx.x * blockDim.x + threadIdx.x;
    if (e >= total) return;
    int s, d;
    if (e < E) { s = ei[e]; d = ei[E + e]; } else { s = e - E; d = s; }
    const float* av = a_src + (size_t)s * HEADS;
    const float* bv = a_dst + (size_t)d * HEADS;
    float* ap = alpha + (size_t)e * HEADS;
    float* mp = m + (size_t)d * HEADS;
#pragma unroll
    for (int h = 0; h < HEADS; ++h) {
        float l = av[h] + bv[h];
        l = (l > 0.0f) ? l : SLOPE * l;      // leaky relu
        ap[h] = l;                            // stash logit in alpha buffer
        atomicMax(&mp[h], l);                 // global_atomic_max_num_f32
    }
}

// -------------- exp(e - m[dst]) + segment-sum into denom -------------------
__global__ __launch_bounds__(256)
void gat_edge_exp(const int* __restrict__ ei, const float* __restrict__ m,
                  float* __restrict__ alpha, float* __restrict__ denom,
                  int E, int total)
{
    const int e = blockIdx.x * blockDim.x + threadIdx.x;
    if (e >= total) return;
    const int d = (e < E) ? ei[E + e] : (e - E);
    const float* mp = m + (size_t)d * HEADS;
    float* ap = alpha + (size_t)e * HEADS;
    float* dp = denom + (size_t)d * HEADS;
#pragma unroll
    for (int h = 0; h < HEADS; ++h) {
        const float ex = __expf(ap[h] - mp[h]);
        ap[h] = ex;
        atomicAdd(&dp[h], ex);                // global_atomic_add_f32
    }
}

// -------- normalize alpha + weighted scatter-add (wave32 per edge) ---------
__global__ __launch_bounds__(256)
void gat_scatter(const int* __restrict__ ei, const float* __restrict__ xh,
                 const float* __restrict__ denom, float* __restrict__ alpha,
                 float* __restrict__ out, int E, int total)
{
    const int gid  = blockIdx.x * blockDim.x + threadIdx.x;
    const int e    = gid >> 5;           // one wave32 per edge
    const int lane = gid & 31;
    if (e >= total) return;
    int s, d;
    if (e < E) { s = ei[e]; d = ei[E + e]; } else { s = e - E; d = s; }

    const int h = lane >> 2;             // lane handles 8 contiguous channels
    const float a = alpha[(size_t)e * HEADS + h] /
                    (denom[(size_t)d * HEADS + h] + 1e-16f);
    if ((lane & 3) == 0)                 // one writer per head: final alpha out
        alpha[(size_t)e * HEADS + h] = a;

    const int off = lane * 8;
    const float4* xv = (const float4*)(xh + (size_t)s * IN_CH + off);
    const float4 v0 = xv[0];
    const float4 v1 = xv[1];
    float* o = out + (size_t)d * IN_CH + off;
    atomicAdd(o + 0, v0.x * a);
    atomicAdd(o + 1, v0.y * a);
    atomicAdd(o + 2, v0.z * a);
    atomicAdd(o + 3, v0.w * a);
    atomicAdd(o + 4, v1.x * a);
    atomicAdd(o + 5, v1.y * a);
    atomicAdd(o + 6, v1.z * a);
    atomicAdd(o + 7, v1.w * a);
}

// ------------------------- bias + residual ---------------------------------
__global__ __launch_bounds__(256)
void gat_bias_res(const float* __restrict__ x, const float* __restrict__ bias,
                  float* __restrict__ out, int n)
{
    const int i = blockIdx.x * blockDim.x + threadIdx.x;   // float4 granularity
    if (i >= n * (IN_CH / 4)) return;
    const float4 xv = ((const float4*)x)[i];
    const float4 bv = ((const float4*)bias)[i & 63];
    float4 ov = ((float4*)out)[i];
    ov.x += xv.x + bv.x;
    ov.y += xv.y + bv.y;
    ov.z += xv.z + bv.z;
    ov.w += xv.w + bv.w;
    ((float4*)out)[i] = ov;
}

// ---------------------------------------------------------------------------
extern "C" void kernel_launch(void* const* d_in, const int* in_sizes, int n_in,
                              void* d_out, int out_size, void* d_ws, size_t ws_size,
                              hipStream_t stream)
{
    const float* x       = (const float*)d_in[0];
    const int*   ei      = (const int*)d_in[1];
    const float* W       = (const float*)d_in[2];
    const float* att_src = (const float*)d_in[3];
    const float* att_dst = (const float*)d_in[4];
    const float* bias    = (const float*)d_in[5];

    const int n     = in_sizes[0] / IN_CH;
    const int E     = in_sizes[1] / 2;
    const int total = E + n;

    float* out   = (float*)d_out;                       // (n, 256)
    float* alpha = out + (size_t)n * IN_CH;             // (E+n, 8)

    char* ws = (char*)d_ws;
    float* xh    = (float*)ws;  ws += (size_t)n * IN_CH * sizeof(float);
    float* a_src = (float*)ws;  ws += (size_t)n * HEADS * sizeof(float);
    float* a_dst = (float*)ws;  ws += (size_t)n * HEADS * sizeof(float);
    float* mbuf  = (float*)ws;  ws += (size_t)n * HEADS * sizeof(float);
    float* denom = (float*)ws;

    gat_init<<<(n * IN_CH + 255) / 256, 256, 0, stream>>>(out, mbuf, denom, n);
    gat_gemm_wmma<<<(n + 15) / 16, 512, 0, stream>>>(x, W, xh, n);
    gat_logits<<<(n * HEADS + 255) / 256, 256, 0, stream>>>(xh, att_src, att_dst,
                                                            a_src, a_dst, n);
    gat_edge_max<<<(total + 255) / 256, 256, 0, stream>>>(ei, a_src, a_dst,
                                                          alpha, mbuf, E, total);
    gat_edge_exp<<<(total + 255) / 256, 256, 0, stream>>>(ei, mbuf, alpha,
                                                          denom, E, total);
    gat_scatter<<<((size_t)total * 32 + 255) / 256, 256, 0, stream>>>(
        ei, xh, denom, alpha, out, E, total);
    gat_bias_res<<<(n * (IN_CH / 4) + 255) / 256, 256, 0, stream>>>(x, bias, out, n);

    (void)n_in; (void)out_size; (void)ws_size;
}